// DenseLayer_68332929679499
// MI455X (gfx1250) — compile-verified
//
#include <hip/hip_runtime.h>

// ---- CDNA5 (gfx1250) wave32 fp32 matvec: out = x @ W + b -----------------
// Memory-bound (256 MB of W @ 23.3 TB/s ~= 11 us floor). Strategy:
//   * split-K: each wave owns (16-column group, K-slice of 1024) -> 4096 waves
//   * W streamed with fully coalesced per-lane b32 loads (consecutive j / lane)
//   * fp32 WMMA (v_wmma_f32_16x16x4_f32) used as the FMA engine:
//       A = x[k0..k0+3] broadcast to all 16 rows, B = 4x16 tile of W
//       -> every row of D is the 16 partial outputs; extract row 0.
//   * partials combined via global_atomic_add_f32 into bias-initialized out.

typedef __attribute__((ext_vector_type(2))) float v2f;
typedef __attribute__((ext_vector_type(8))) float v8f;

#define IN_LEN            8192
#define OUT_LEN           8192
#define KSPLIT            8
#define KSLICE            (IN_LEN / KSPLIT)          // 1024
#define COLS_PER_WAVE     16
#define NCOLGRP           (OUT_LEN / COLS_PER_WAVE)  // 512
#define WAVES_PER_BLOCK   8
#define THREADS_PER_BLOCK (WAVES_PER_BLOCK * 32)     // 256 threads = 8 waves
#define MATVEC_BLOCKS     ((NCOLGRP * KSPLIT) / WAVES_PER_BLOCK) // 512

__global__ __launch_bounds__(THREADS_PER_BLOCK)
void bias_init_kernel(const float* __restrict__ b, float* __restrict__ out) {
    int i = blockIdx.x * blockDim.x + threadIdx.x;
    if (i < OUT_LEN) out[i] = b[i];
}

__global__ __launch_bounds__(THREADS_PER_BLOCK)
void matvec_wmma_f32_kernel(const float* __restrict__ x,
                            const float* __restrict__ W,
                            float* __restrict__ out) {
    const int lane        = threadIdx.x & 31;
    const int waveInBlock = threadIdx.x >> 5;
    const int flatWave    = blockIdx.x * WAVES_PER_BLOCK + waveInBlock;

    const int slice    = flatWave >> 9;              // 0..KSPLIT-1  (512 groups/slice)
    const int colGroup = flatWave & (NCOLGRP - 1);   // 0..511
    const int j0       = colGroup * COLS_PER_WAVE;
    const int kBeg     = slice * KSLICE;

    // 16x4 f32 A-matrix layout: lanes 0-15 hold K=0 (V0), K=1 (V1);
    // lanes 16-31 hold K=2 (V0), K=3 (V1).  B (4x16): row K striped across
    // the matching lane half, N = laneLo (16 consecutive columns -> coalesced).
    const int laneLo = lane & 15;
    const int rowOff = (lane >> 4) << 1;             // 0 or 2

    const float* wp = W + (size_t)(kBeg + rowOff) * OUT_LEN + (size_t)(j0 + laneLo);
    const float* xp = x + kBeg + rowOff;

    v8f acc0 = {};
    v8f acc1 = {};

    // KSLICE=1024 k-values, 4 per WMMA, 2 WMMAs per iter -> 128 iters
    #pragma unroll 2
    for (int it = 0; it < KSLICE / 8; ++it) {
        // x chunk (broadcast A): one 8B load per lane, hits cache
        v2f a0 = *(const v2f*)(xp);
        v2f a1 = *(const v2f*)(xp + 4);

        // W 4x16 tiles (B): two b32 loads per lane per WMMA, wave-coalesced
        v2f b0;
        b0.x = wp[0];
        b0.y = wp[OUT_LEN];
        v2f b1;
        b1.x = wp[(size_t)4 * OUT_LEN];
        b1.y = wp[(size_t)5 * OUT_LEN];

        // keep L2 -> WGP pipe primed (emits global_prefetch_b8)
        __builtin_prefetch(wp + (size_t)32 * OUT_LEN, 0, 1);

        // D = A*B + C  (fp32, full precision); two accumulators break the
        // D->C serial dependency chain
        acc0 = __builtin_amdgcn_wmma_f32_16x16x4_f32(
            false, a0, false, b0, (short)0, acc0, false, false);
        acc1 = __builtin_amdgcn_wmma_f32_16x16x4_f32(
            false, a1, false, b1, (short)0, acc1, false, false);

        wp += (size_t)8 * OUT_LEN;
        xp += 8;
    }

    // All 16 rows of D are identical; row 0 = element 0 of v8f, lanes 0..15,
    // N = laneLo.  Lanes 16..31 hold duplicates (row M=8) -> predicate off.
    float total = acc0[0] + acc1[0];
    if (lane < 16) {
        unsafeAtomicAdd(&out[j0 + laneLo], total);   // global_atomic_add_f32
    }
}

extern "C" void kernel_launch(void* const* d_in, const int* in_sizes, int n_in,
                              void* d_out, int out_size, void* d_ws, size_t ws_size,
                              hipStream_t stream) {
    (void)in_sizes; (void)n_in; (void)d_ws; (void)ws_size; (void)out_size;

    const float* x = (const float*)d_in[0];   // (1, 8192) f32
    const float* W = (const float*)d_in[1];   // (8192, 8192) f32 row-major
    const float* b = (const float*)d_in[2];   // (8192,) f32
    float*     out = (float*)d_out;           // (1, 8192) f32

    // 1) out = bias
    bias_init_kernel<<<OUT_LEN / THREADS_PER_BLOCK, THREADS_PER_BLOCK, 0, stream>>>(b, out);

    // 2) out += x @ W  (split-K partials via fp32 global atomics)
    matvec_wmma_f32_kernel<<<MATVEC_BLOCKS, THREADS_PER_BLOCK, 0, stream>>>(x, W, out);
}